// OurLoss_40269613368003
// MI455X (gfx1250) — compile-verified
//
#include <hip/hip_runtime.h>
#include <hip/hip_bf16.h>

typedef __attribute__((ext_vector_type(2))) float v2f;
typedef __attribute__((ext_vector_type(8))) float v8f;

#define BT   128   // tracklets
#define SS   32    // samples per tracklet
#define DD   128   // feature dim
#define LSTR 132   // padded LDS row stride (floats) -> conflict-free wmma operand reads
#define TSTR 33    // padded 32x32 stats tile stride

// ---------------- wave helpers (wave32) ----------------
__device__ __forceinline__ float wave_max32(float v) {
#pragma unroll
  for (int off = 16; off > 0; off >>= 1) v = fmaxf(v, __shfl_xor(v, off, 32));
  return v;
}
__device__ __forceinline__ float wave_sum32(float v) {
#pragma unroll
  for (int off = 16; off > 0; off >>= 1) v += __shfl_xor(v, off, 32);
  return v;
}

// exact 3rd- and 6th-largest of the 32 per-lane values (duplicates handled:
// remove exactly one instance per extraction round, like top_k)
__device__ __forceinline__ void topk_3_6(float v, float& t3, float& t6) {
  float x = v;
  t3 = 0.0f; t6 = 0.0f;
  for (int it = 1; it <= 6; ++it) {
    float m = wave_max32(x);
    if (it == 3) t3 = m;
    if (it == 6) t6 = m;
    unsigned long long bal = __ballot(x == m);
    int first = (int)__builtin_ctzll(bal);
    if ((int)(threadIdx.x & 31) == first) x = -INFINITY;
  }
}

// ---------------- 32x32 Gram tile via V_WMMA_F32_16X16X4_F32 ----------------
// A, Bm: LDS, 32 rows x LSTR floats (row-major, 128 valid cols).
// Sm: LDS 32 x TSTR output, Sm[r][c] = dot(A_row_r, B_row_c).
// One full wave32, EXEC all-ones (required for WMMA).
__device__ __forceinline__ void gemm32_wmma(const float* A, const float* Bm, float* Sm) {
  const int lane = threadIdx.x & 31;
  const int half = lane >> 4;   // 0: K={0,1}, 1: K={2,3}  (per ISA A/B f32 layout)
  const int lm   = lane & 15;   // M (for A) / N (for B) within 16-wide tile
  v8f acc00 = {}, acc01 = {}, acc10 = {}, acc11 = {};
  for (int k = 0; k < DD; k += 4) {
    const int kk = k + 2 * half;
    v2f a0 = { A[lm * LSTR + kk],        A[lm * LSTR + kk + 1] };
    v2f a1 = { A[(lm + 16) * LSTR + kk], A[(lm + 16) * LSTR + kk + 1] };
    v2f b0 = { Bm[lm * LSTR + kk],        Bm[lm * LSTR + kk + 1] };
    v2f b1 = { Bm[(lm + 16) * LSTR + kk], Bm[(lm + 16) * LSTR + kk + 1] };
    acc00 = __builtin_amdgcn_wmma_f32_16x16x4_f32(false, a0, false, b0, (short)0, acc00, false, false);
    acc01 = __builtin_amdgcn_wmma_f32_16x16x4_f32(false, a0, false, b1, (short)0, acc01, false, false);
    acc10 = __builtin_amdgcn_wmma_f32_16x16x4_f32(false, a1, false, b0, (short)0, acc10, false, false);
    acc11 = __builtin_amdgcn_wmma_f32_16x16x4_f32(false, a1, false, b1, (short)0, acc11, false, false);
  }
  // C/D layout: VGPR v -> row M = v + 8*(lane/16), col N = lane%16
#pragma unroll
  for (int v = 0; v < 8; ++v) {
    const int r0 = v + 8 * half;
    Sm[r0 * TSTR + lm]              = acc00[v];
    Sm[r0 * TSTR + (lm + 16)]       = acc01[v];
    Sm[(r0 + 16) * TSTR + lm]       = acc10[v];
    Sm[(r0 + 16) * TSTR + (lm + 16)] = acc11[v];
  }
}

__device__ __forceinline__ float dist_exp(float s) {
  // exp(sqrt(clip(2 - 2*s, 1e-12)))  for unit-norm features
  return expf(sqrtf(fmaxf(2.0f - 2.0f * s, 1e-12f)));
}

// ---------------- kernel 1: L2 normalize each (tracklet,sample) vector ------
__global__ __launch_bounds__(32) void norm_kernel(const float* __restrict__ X,
                                                  float* __restrict__ Xn) {
  const int vec  = blockIdx.x;          // 0..4095
  const int lane = threadIdx.x;         // 0..31, 4 floats each
  float4 v = ((const float4*)(X + (size_t)vec * DD))[lane];
  float ss = v.x * v.x + v.y * v.y + v.z * v.z + v.w * v.w;
  ss = wave_sum32(ss);
  const float scale = 1.0f / fmaxf(sqrtf(ss), 1e-12f);
  v.x *= scale; v.y *= scale; v.z *= scale; v.w *= scale;
  ((float4*)(Xn + (size_t)vec * DD))[lane] = v;
}

// ---------------- kernel 2: per-tracklet self exp-dist row sums -------------
__global__ __launch_bounds__(32) void selfsum_kernel(const float* __restrict__ Xn,
                                                     float* __restrict__ selfS) {
  __shared__ float lA[SS * LSTR];
  __shared__ float lS[SS * TSTR];
  const int i = blockIdx.x;
  const int lane = threadIdx.x;
  for (int r = 0; r < SS; ++r)
    *(float4*)&lA[r * LSTR + lane * 4] =
        ((const float4*)(Xn + ((size_t)i * SS + r) * DD))[lane];
  __syncthreads();
  gemm32_wmma(lA, lA, lS);
  __syncthreads();
  float rsum = 0.0f;
  for (int c = 0; c < SS; ++c) rsum += dist_exp(lS[lane * TSTR + c]);
  selfS[i * SS + lane] = rsum;
}

// ---------------- kernel 3: per-pair stats -> D matrix ----------------------
__global__ __launch_bounds__(32) void pair_kernel(const float* __restrict__ Xn,
                                                  const float* __restrict__ selfS,
                                                  const int* __restrict__ tg,
                                                  float* __restrict__ Dm) {
  __shared__ float lA[SS * LSTR];
  __shared__ float lB[SS * LSTR];
  __shared__ float lS[SS * TSTR];
  const int i = blockIdx.x, j = blockIdx.y;
  const int lane = threadIdx.x;
  if (i == j) {                          // reference forces diag to 0
    if (lane == 0) Dm[i * BT + j] = 0.0f;
    return;
  }
  for (int r = 0; r < SS; ++r) {
    *(float4*)&lA[r * LSTR + lane * 4] =
        ((const float4*)(Xn + ((size_t)i * SS + r) * DD))[lane];
    *(float4*)&lB[r * LSTR + lane * 4] =
        ((const float4*)(Xn + ((size_t)j * SS + r) * DD))[lane];
  }
  __syncthreads();
  gemm32_wmma(lA, lB, lS);               // lS[r][c] = <xi_r, xj_c>
  __syncthreads();
  // row pass: thread t owns row t; transform Gram -> exp(dist) in place
  float rsum = 0.0f, rmin = INFINITY;
  for (int c = 0; c < SS; ++c) {
    float e = dist_exp(lS[lane * TSTR + c]);
    lS[lane * TSTR + c] = e;
    rsum += e;
    rmin = fminf(rmin, e);
  }
  __syncthreads();
  // col pass: thread t owns column t
  float csum = 0.0f, cmn = INFINITY;
  for (int r = 0; r < SS; ++r) {
    float e = lS[r * TSTR + lane];
    csum += e;
    cmn = fminf(cmn, e);
  }
  const float av = rmin / (selfS[i * SS + lane] + rsum);  // row-softmax slice min
  const float cv = cmn / (selfS[j * SS + lane] + csum);   // col-softmax slice min
  float a3, a6, c3, c6;
  topk_3_6(av, a3, a6);
  topk_3_6(cv, c3, c6);
  if (lane == 0) {
    const float vsame = fmaxf(a6, c6);   // k_ap-th largest
    const float vdiff = fminf(a3, c3);   // k_an-th largest
    Dm[i * BT + j] = (tg[i] == tg[j]) ? vsame : vdiff;
  }
}

// ---------------- kernel 4: hardest pos/neg margin loss ---------------------
__global__ __launch_bounds__(128) void loss_kernel(const float* __restrict__ Dm,
                                                   const int* __restrict__ tg,
                                                   float* __restrict__ out) {
  __shared__ float red[BT];
  const int i = threadIdx.x;
  const int ti = tg[i];
  float ap = -INFINITY, an = INFINITY;
  for (int j = 0; j < BT; ++j) {
    const float d = Dm[i * BT + j];      // d(i,i)==0 counts toward positives
    if (tg[j] == ti) ap = fmaxf(ap, d);
    else             an = fminf(an, d);
  }
  red[i] = fmaxf(ap - an + 0.3f, 0.0f);
  __syncthreads();
  for (int s2 = 64; s2 > 0; s2 >>= 1) {
    if (i < s2) red[i] += red[i + s2];
    __syncthreads();
  }
  if (i == 0) out[0] = red[0] * (1.0f / (float)BT);
}

// ---------------- launch ----------------------------------------------------
extern "C" void kernel_launch(void* const* d_in, const int* in_sizes, int n_in,
                              void* d_out, int out_size, void* d_ws, size_t ws_size,
                              hipStream_t stream) {
  const float* X  = (const float*)d_in[0];   // (128,32,128) f32
  const int*   tg = (const int*)d_in[1];     // (128,) int
  float* out = (float*)d_out;
  char* ws = (char*)d_ws;
  float* Xn    = (float*)ws;                                              // 2 MB
  float* selfS = (float*)(ws + (size_t)BT * SS * DD * sizeof(float));     // 16 KB
  float* Dm    = (float*)(ws + (size_t)BT * SS * DD * sizeof(float)
                             + (size_t)BT * SS * sizeof(float));          // 64 KB

  norm_kernel<<<BT * SS, 32, 0, stream>>>(X, Xn);
  selfsum_kernel<<<BT, 32, 0, stream>>>(Xn, selfS);
  pair_kernel<<<dim3(BT, BT), 32, 0, stream>>>(Xn, selfS, tg, Dm);
  loss_kernel<<<1, BT, 0, stream>>>(Dm, tg, out);
}